// ClsFreeRPN_18090402250922
// MI455X (gfx1250) — compile-verified
//
#include <hip/hip_runtime.h>

#define NCAT   4768
#define NWORDS 149        // 4768 / 32
#define NTILER 298        // 4768 / 16
#define POSTK  1000
#define IMGSZ  1024.0f
#define NMS_T  0.7f
#define BATCH  8

typedef __attribute__((ext_vector_type(2))) float v2f;
typedef __attribute__((ext_vector_type(8))) float v8f;

__device__ __forceinline__ float NEGINF() { return -__builtin_inff(); }

// monotone float -> uint key (larger float => larger key)
__device__ __forceinline__ unsigned fkey(float f) {
  unsigned u = __builtin_bit_cast(unsigned, f);
  return (u & 0x80000000u) ? ~u : (u | 0x80000000u);
}

// wave32 lane broadcast via ds_bpermute (index may vary per lane)
__device__ __forceinline__ float bperm(float v, int srcLane) {
  return __builtin_bit_cast(
      float, __builtin_amdgcn_ds_bpermute(srcLane << 2, __builtin_bit_cast(int, v)));
}

// ---------------------------------------------------------------------------
// K1: per-(image,level) radix-select top-K on raw ctr scores, then decode the
// selected boxes (Box2BoxTransformLinear), clip, validity-mask score.
// ---------------------------------------------------------------------------
__device__ __forceinline__ void decode_store(
    const float* __restrict__ deltas, const float* __restrict__ ctr,
    const float* __restrict__ anchors, int img, int N, int i, int slot,
    float* __restrict__ cs, float* __restrict__ cb) {
  const float* a4 = anchors + (size_t)i * 4;
  const float* d4 = deltas + ((size_t)img * N + i) * 4;
  float x1 = a4[0], y1 = a4[1], x2 = a4[2], y2 = a4[3];
  float w = x2 - x1, h = y2 - y1;
  float cx = 0.5f * (x1 + x2), cy = 0.5f * (y1 + y2);
  float bx1 = cx - d4[0] * w, by1 = cy - d4[1] * h;
  float bx2 = cx + d4[2] * w, by2 = cy + d4[3] * h;
  bx1 = fminf(fmaxf(bx1, 0.f), IMGSZ);
  by1 = fminf(fmaxf(by1, 0.f), IMGSZ);
  bx2 = fminf(fmaxf(bx2, 0.f), IMGSZ);
  by2 = fminf(fmaxf(by2, 0.f), IMGSZ);
  bool valid = (bx2 - bx1 > 0.f) && (by2 - by1 > 0.f);
  float s = valid ? ctr[(size_t)img * N + i] : NEGINF();
  size_t base = (size_t)img * NCAT + slot;
  cs[base] = s;
  cb[base * 4 + 0] = bx1;
  cb[base * 4 + 1] = by1;
  cb[base * 4 + 2] = bx2;
  cb[base * 4 + 3] = by2;
}

__global__ __launch_bounds__(256) void topk_decode_kernel(
    const float* __restrict__ deltas, const float* __restrict__ ctr,
    const float* __restrict__ anchors, int N, int K, int lvloff,
    float* __restrict__ cs, float* __restrict__ cb) {
  __shared__ unsigned hist[256];
  __shared__ unsigned sh_pref, sh_rem, sh_cnt, sh_ngt;
  int img = blockIdx.x, tid = threadIdx.x;
  const float* C = ctr + (size_t)img * N;
  if (tid == 0) { sh_pref = 0u; sh_rem = (unsigned)K; }
  __syncthreads();
  // 4-pass (8 bits each) radix select of the K-th largest key
  for (int p = 3; p >= 0; --p) {
    int shift = p * 8;
    hist[tid] = 0u;
    __syncthreads();
    unsigned pref = sh_pref;
    for (int i = tid; i < N; i += 256) {
      if (i + 2048 < N) __builtin_prefetch(&C[i + 2048], 0, 0);
      unsigned u = fkey(C[i]);
      bool inpref = (p == 3) || ((u >> (shift + 8)) == pref);
      if (inpref) atomicAdd(&hist[(u >> shift) & 255u], 1u);
    }
    __syncthreads();
    if (tid == 0) {
      unsigned rem = sh_rem, acc = 0u, sel = 0u;
      for (int bin = 255; bin >= 0; --bin) {
        acc += hist[bin];
        if (acc >= rem) { sel = (unsigned)bin; sh_rem = rem - (acc - hist[bin]); break; }
      }
      sh_pref = (pref << 8) | sel;
    }
    __syncthreads();
  }
  unsigned T = sh_pref;  // exact key of K-th largest
  if (tid == 0) sh_cnt = 0u;
  __syncthreads();
  for (int i = tid; i < N; i += 256) {           // strictly greater
    unsigned u = fkey(C[i]);
    if (u > T) {
      unsigned pos = atomicAdd(&sh_cnt, 1u);
      decode_store(deltas, ctr, anchors, img, N, i, lvloff + (int)pos, cs, cb);
    }
  }
  __syncthreads();
  if (tid == 0) { sh_ngt = sh_cnt; sh_cnt = 0u; }
  __syncthreads();
  unsigned ngt = sh_ngt;
  for (int i = tid; i < N; i += 256) {           // ties at threshold fill remainder
    unsigned u = fkey(C[i]);
    if (u == T) {
      unsigned t = atomicAdd(&sh_cnt, 1u);
      unsigned pos = ngt + t;
      if (pos < (unsigned)K)
        decode_store(deltas, ctr, anchors, img, N, i, lvloff + (int)pos, cs, cb);
    }
  }
}

// ---------------------------------------------------------------------------
// K2: per-image bitonic sort of 4768 (score, slot) pairs (padded to 8192) in
// LDS (64 KB of the 320 KB/WGP), descending score, tie-break ascending index
// (matches stable argsort(-scores)). Then gather sorted boxes and the
// per-level-offset boxes used for batched NMS.
// ---------------------------------------------------------------------------
__global__ __launch_bounds__(1024) void sort_gather_kernel(
    const float* __restrict__ cs, const float* __restrict__ cb,
    float* __restrict__ ss, float* __restrict__ sb, float* __restrict__ sbo) {
  __shared__ float kk[8192];
  __shared__ int vv[8192];
  int img = blockIdx.x, tid = threadIdx.x;
  for (int i = tid; i < 8192; i += 1024) {
    kk[i] = (i < NCAT) ? cs[(size_t)img * NCAT + i] : NEGINF();
    vv[i] = i;
  }
  __syncthreads();
  for (unsigned size = 2; size <= 8192u; size <<= 1) {
    for (unsigned stride = size >> 1; stride > 0; stride >>= 1) {
      for (unsigned t = tid; t < 4096u; t += 1024u) {
        unsigned lo = ((t & ~(stride - 1u)) << 1) | (t & (stride - 1u));
        unsigned hi = lo + stride;
        bool dirAsc = ((lo & size) == 0u);
        float ka = kk[lo], kb = kk[hi];
        int va = vv[lo], vb = vv[hi];
        bool before = (ka > kb) || (ka == kb && va < vb);
        if (dirAsc ? !before : before) {
          kk[lo] = kb; kk[hi] = ka; vv[lo] = vb; vv[hi] = va;
        }
      }
      __syncthreads();
    }
  }
  for (int i = tid; i < NCAT; i += 1024) {
    float s = kk[i];
    int sl = vv[i];
    ss[(size_t)img * NCAT + i] = s;
    int lvl = sl < 1000 ? 0 : sl < 2000 ? 1 : sl < 3000 ? 2 : sl < 4000 ? 3 : 4;
    float off = (float)lvl * (IMGSZ + 1.0f);
    size_t src = ((size_t)img * NCAT + sl) * 4;
    size_t dst = ((size_t)img * NCAT + i) * 4;
#pragma unroll
    for (int c = 0; c < 4; ++c) {
      float v = cb[src + c];
      sb[dst + c] = v;
      sbo[dst + c] = v + off;
    }
  }
}

// ---------------------------------------------------------------------------
// K3: IoU suppression mask. One wave32 per 16-row x 32-col tile. The pairwise
// area sum (IoU union term) is a rank-2 outer product computed with
// V_WMMA_F32_16X16X4_F32:  A = [area_row, 1, 0, 0], B = [1; area_col; 0; 0]
// => D[m][n] = area_row[m] + area_col[n]. Intersections via VALU min/max in
// the same C/D lane layout; wave32 ballots assemble 32-bit mask words.
// ---------------------------------------------------------------------------
__global__ __launch_bounds__(32) void iou_mask_kernel(
    const float* __restrict__ sbo, unsigned* __restrict__ mask) {
  int img = blockIdx.y;
  int lin = blockIdx.x;
  int ti = lin / NWORDS, wj = lin % NWORDS;
  int lane = threadIdx.x;
  int i0 = ti * 16, c0 = wj * 32;
  const float* P = sbo + (size_t)img * NCAT * 4;

  int rr = i0 + (lane & 15);
  float rx1 = P[rr * 4 + 0], ry1 = P[rr * 4 + 1], rx2 = P[rr * 4 + 2], ry2 = P[rr * 4 + 3];
  float rA = fmaxf(rx2 - rx1, 0.f) * fmaxf(ry2 - ry1, 0.f);
  int cc = c0 + lane;
  float cx1 = P[cc * 4 + 0], cy1 = P[cc * 4 + 1], cx2 = P[cc * 4 + 2], cy2 = P[cc * 4 + 3];
  float cA = fmaxf(cx2 - cx1, 0.f) * fmaxf(cy2 - cy1, 0.f);

  bool lo16 = lane < 16;
  // A (16x4): lanes 0-15 -> K0,K1 ; lanes 16-31 -> K2,K3 (zero)
  v2f a;  a.x = lo16 ? rA : 0.f;  a.y = lo16 ? 1.f : 0.f;
  // B (4x16): lanes 0-15 -> K0=1, K1=area_col ; lanes 16-31 -> K2,K3 (zero)
  v2f b0; b0.x = lo16 ? 1.f : 0.f; b0.y = lo16 ? cA : 0.f;
  float cAhi = bperm(cA, (lane & 15) + 16);
  v2f b1; b1.x = lo16 ? 1.f : 0.f; b1.y = lo16 ? cAhi : 0.f;
  v8f z = {0.f, 0.f, 0.f, 0.f, 0.f, 0.f, 0.f, 0.f};
  v8f d0 = __builtin_amdgcn_wmma_f32_16x16x4_f32(false, a, false, b0, (short)0, z, false, false);
  v8f d1 = __builtin_amdgcn_wmma_f32_16x16x4_f32(false, a, false, b1, (short)0, z, false, false);

  int nl = lane & 15;
  float Ax1 = bperm(cx1, nl),      Ay1 = bperm(cy1, nl);
  float Ax2 = bperm(cx2, nl),      Ay2 = bperm(cy2, nl);
  float Bx1 = bperm(cx1, nl + 16), By1 = bperm(cy1, nl + 16);
  float Bx2 = bperm(cx2, nl + 16), By2 = bperm(cy2, nl + 16);
  int rowAdd = (lane >= 16) ? 8 : 0;
  unsigned myword = 0u;
#pragma unroll
  for (int k2 = 0; k2 < 8; ++k2) {
    int m = k2 + rowAdd;
    float Rx1 = bperm(rx1, m), Ry1 = bperm(ry1, m);
    float Rx2 = bperm(rx2, m), Ry2 = bperm(ry2, m);
    int gi = i0 + m;
    float iw0 = fmaxf(fminf(Rx2, Ax2) - fmaxf(Rx1, Ax1), 0.f);
    float ih0 = fmaxf(fminf(Ry2, Ay2) - fmaxf(Ry1, Ay1), 0.f);
    float inter0 = iw0 * ih0;
    float un0 = d0[k2] - inter0 + 1e-8f;          // area_i + area_j - inter + eps
    bool p0 = (inter0 > NMS_T * un0) && ((c0 + nl) > gi);
    float iw1 = fmaxf(fminf(Rx2, Bx2) - fmaxf(Rx1, Bx1), 0.f);
    float ih1 = fmaxf(fminf(Ry2, By2) - fmaxf(Ry1, By1), 0.f);
    float inter1 = iw1 * ih1;
    float un1 = d1[k2] - inter1 + 1e-8f;
    bool p1 = (inter1 > NMS_T * un1) && ((c0 + 16 + nl) > gi);
    unsigned bal0 = __builtin_amdgcn_ballot_w32(p0);
    unsigned bal1 = __builtin_amdgcn_ballot_w32(p1);
    unsigned wlow  = (bal0 & 0xffffu) | (bal1 << 16);         // row k2
    unsigned whigh = (bal0 >> 16) | (bal1 & 0xffff0000u);     // row k2+8
    if (lane == k2)      myword = wlow;
    if (lane == k2 + 16) myword = whigh;
  }
  if ((lane & 15) < 8) {
    int row = i0 + (lane & 7) + rowAdd;
    mask[((size_t)img * NCAT + row) * NWORDS + wj] = myword;
  }
}

// ---------------------------------------------------------------------------
// K4: greedy scan (sequential, one wave32 per image; removed-bitmap in LDS,
// next-row prefetch via global_prefetch_b8), then ballot-compacted writeout:
// kept entries first (score order), then suppressed/invalid with -inf.
// ---------------------------------------------------------------------------
__global__ __launch_bounds__(32) void nms_scan_out_kernel(
    const float* __restrict__ ss, const float* __restrict__ sb,
    const unsigned* __restrict__ mask, float* __restrict__ out) {
  __shared__ unsigned rem[NWORDS];
  int img = blockIdx.x, lane = threadIdx.x;
  for (int t = lane; t < NWORDS; t += 32) rem[t] = 0u;
  __syncthreads();
  const float* S = ss + (size_t)img * NCAT;
  const unsigned* M = mask + (size_t)img * NCAT * NWORDS;
  for (int i = 0; i < NCAT; ++i) {
    if (i + 1 < NCAT)
      __builtin_prefetch(M + (size_t)(i + 1) * NWORDS + lane * 8, 0, 0);
    unsigned w = rem[i >> 5];
    float s = S[i];
    if ((s > NEGINF()) && !((w >> (i & 31)) & 1u)) {
      const unsigned* row = M + (size_t)i * NWORDS;
      for (int t = lane; t < NWORDS; t += 32) rem[t] |= row[t];
    }
  }
  __syncthreads();
  const float* B4 = sb + (size_t)img * NCAT * 4;
  float* O = out + (size_t)img * POSTK * 5;
  unsigned lmask = (1u << lane) - 1u;
  int outp = 0;
  for (int base = 0; base < NCAT; base += 32) {   // kept, in score order
    int i = base + lane;
    unsigned w = rem[i >> 5];
    float s = S[i];
    bool kept = (s > NEGINF()) && !((w >> (i & 31)) & 1u);
    unsigned mb = __builtin_amdgcn_ballot_w32(kept);
    int pos = outp + __popc(mb & lmask);
    if (kept && pos < POSTK) {
      O[pos * 5 + 0] = B4[i * 4 + 0];
      O[pos * 5 + 1] = B4[i * 4 + 1];
      O[pos * 5 + 2] = B4[i * 4 + 2];
      O[pos * 5 + 3] = B4[i * 4 + 3];
      O[pos * 5 + 4] = s;
    }
    outp += __popc(mb);
  }
  for (int base = 0; base < NCAT; base += 32) {   // non-kept filler with -inf
    int i = base + lane;
    unsigned w = rem[i >> 5];
    float s = S[i];
    bool rej = !((s > NEGINF()) && !((w >> (i & 31)) & 1u));
    unsigned mb = __builtin_amdgcn_ballot_w32(rej);
    int pos = outp + __popc(mb & lmask);
    if (rej && pos < POSTK) {
      O[pos * 5 + 0] = B4[i * 4 + 0];
      O[pos * 5 + 1] = B4[i * 4 + 1];
      O[pos * 5 + 2] = B4[i * 4 + 2];
      O[pos * 5 + 3] = B4[i * 4 + 3];
      O[pos * 5 + 4] = NEGINF();
    }
    outp += __popc(mb);
  }
}

// ---------------------------------------------------------------------------
extern "C" void kernel_launch(void* const* d_in, const int* in_sizes, int n_in,
                              void* d_out, int out_size, void* d_ws, size_t ws_size,
                              hipStream_t stream) {
  (void)in_sizes; (void)n_in; (void)out_size; (void)ws_size;
  static const int Ns[5]   = {196608, 49152, 12288, 3072, 768};
  static const int Ks[5]   = {1000, 1000, 1000, 1000, 768};
  static const int Offs[5] = {0, 1000, 2000, 3000, 4000};
  // workspace layout (~25 MB)
  float* cs  = (float*)d_ws;                         // [B][4768] candidate scores
  float* cb  = cs + (size_t)BATCH * NCAT;            // [B][4768][4] candidate boxes
  float* ss  = cb + (size_t)BATCH * NCAT * 4;        // [B][4768] sorted scores
  float* sb  = ss + (size_t)BATCH * NCAT;            // [B][4768][4] sorted boxes (clipped)
  float* sbo = sb + (size_t)BATCH * NCAT * 4;        // [B][4768][4] level-offset boxes
  unsigned* mask = (unsigned*)(sbo + (size_t)BATCH * NCAT * 4);  // [B][4768][149]

  // inputs are interleaved per level: deltas, ctr, anchors
  for (int l = 0; l < 5; ++l) {
    topk_decode_kernel<<<dim3(BATCH), dim3(256), 0, stream>>>(
        (const float*)d_in[3 * l + 0], (const float*)d_in[3 * l + 1],
        (const float*)d_in[3 * l + 2], Ns[l], Ks[l], Offs[l], cs, cb);
  }
  sort_gather_kernel<<<dim3(BATCH), dim3(1024), 0, stream>>>(cs, cb, ss, sb, sbo);
  iou_mask_kernel<<<dim3(NTILER * NWORDS, BATCH), dim3(32), 0, stream>>>(sbo, mask);
  nms_scan_out_kernel<<<dim3(BATCH), dim3(32), 0, stream>>>(ss, sb, mask, (float*)d_out);
}